// Generator1_72971494359047
// MI455X (gfx1250) — compile-verified
//
#include <hip/hip_runtime.h>

typedef __attribute__((ext_vector_type(2))) float v2f;
typedef __attribute__((ext_vector_type(8))) float v8f;

#define IN_FEAT 64
#define HIDDEN 128
#define OUT_FEAT 64

// ---------- monotonic float <-> uint encoding for atomic segment-max ----------
__device__ __forceinline__ unsigned enc_f32(float f) {
  unsigned u = __float_as_uint(f);
  return (u & 0x80000000u) ? ~u : (u | 0x80000000u);
}
__device__ __forceinline__ float dec_f32(unsigned k) {
  unsigned u = (k & 0x80000000u) ? (k & 0x7fffffffu) : ~k;
  return __uint_as_float(u);
}
__device__ __forceinline__ float lrelu02(float v) { return v > 0.f ? v : 0.2f * v; }

// ---------- init kernels ----------
__global__ void init_nodes_kernel(float* deg, unsigned* menc, float* denom, int n) {
  int i = blockIdx.x * blockDim.x + threadIdx.x;
  if (i < n) { deg[i] = 1.0f; menc[i] = 0u; denom[i] = 0.0f; }
}

__global__ void init_bias_kernel(float* buf, const float* __restrict__ bias,
                                 int total, int featmask) {
  int i = blockIdx.x * blockDim.x + threadIdx.x;
  if (i < total) buf[i] = bias[i & featmask];
}

// ---------- degree / rsqrt ----------
__global__ void deg_kernel(const int* __restrict__ dst, float* deg, int E) {
  int e = blockIdx.x * blockDim.x + threadIdx.x;
  if (e < E) atomicAdd(&deg[dst[e]], 1.0f);
}
__global__ void dinv_kernel(const float* __restrict__ deg, float* dinv, int n) {
  int i = blockIdx.x * blockDim.x + threadIdx.x;
  if (i < n) dinv[i] = rsqrtf(deg[i]);
}

// ---------- GEMM1: hx = z @ w1   (N x 64) @ (64 x 128), f32 WMMA ----------
__global__ __launch_bounds__(256) void gemm1_wmma_kernel(
    const float* __restrict__ A, const float* __restrict__ B,
    float* __restrict__ C, int Nn) {
  const int wave = threadIdx.x >> 5;       // 8 waves -> 8 N-tiles (128 cols)
  const int lane = threadIdx.x & 31;
  const int half = lane >> 4;
  const int r = lane & 15;
  const int m0 = blockIdx.x * 16;
  const int n0 = wave * 16;
  int mrow = m0 + r; if (mrow >= Nn) mrow = Nn - 1;   // clamp A loads for tail
  v8f c = {};
#pragma unroll
  for (int k0 = 0; k0 < IN_FEAT; k0 += 4) {
    const int ka = k0 + 2 * half;
    v2f a, b;
    a.x = A[(size_t)mrow * IN_FEAT + ka];
    a.y = A[(size_t)mrow * IN_FEAT + ka + 1];
    b.x = B[ka * HIDDEN + n0 + r];
    b.y = B[(ka + 1) * HIDDEN + n0 + r];
    c = __builtin_amdgcn_wmma_f32_16x16x4_f32(false, a, false, b, (short)0, c,
                                              false, false);
  }
#pragma unroll
  for (int v = 0; v < 8; ++v) {
    int row = m0 + v + 8 * half;
    if (row < Nn) C[(size_t)row * HIDDEN + n0 + r] = c[v];
  }
}

// ---------- GCN scatter: h1[dst] += dinv[src]*dinv[dst] * hx[src] ----------
__global__ void gcn_scatter_kernel(const int* __restrict__ src,
                                   const int* __restrict__ dst,
                                   const float* __restrict__ dinv,
                                   const float* __restrict__ hx,
                                   float* h1, int E, int Nn) {
  int wid = (blockIdx.x * blockDim.x + threadIdx.x) >> 5;  // one wave per edge
  int lane = threadIdx.x & 31;
  if (wid >= E + Nn) return;
  int s, d;
  if (wid < E) { s = src[wid]; d = dst[wid]; } else { s = d = wid - E; }
  float norm = dinv[s] * dinv[d];
  const float4 v = *(const float4*)(hx + (size_t)s * HIDDEN + lane * 4);
  float* o = h1 + (size_t)d * HIDDEN + lane * 4;
  atomicAdd(o + 0, norm * v.x);
  atomicAdd(o + 1, norm * v.y);
  atomicAdd(o + 2, norm * v.z);
  atomicAdd(o + 3, norm * v.w);
}

// ---------- GEMM2: xl = PReLU(h1) @ w2  (N x 128) @ (128 x 64), f32 WMMA ----------
__global__ __launch_bounds__(128) void gemm2_wmma_kernel(
    const float* __restrict__ H, const float* __restrict__ PW,
    const float* __restrict__ B, float* __restrict__ XL, int Nn) {
  const int wave = threadIdx.x >> 5;       // 4 waves -> 4 N-tiles (64 cols)
  const int lane = threadIdx.x & 31;
  const int half = lane >> 4;
  const int r = lane & 15;
  const int m0 = blockIdx.x * 16;
  const int n0 = wave * 16;
  int mrow = m0 + r; if (mrow >= Nn) mrow = Nn - 1;
  v8f c = {};
#pragma unroll
  for (int k0 = 0; k0 < HIDDEN; k0 += 4) {
    const int ka = k0 + 2 * half;
    float h0 = H[(size_t)mrow * HIDDEN + ka];
    float h1v = H[(size_t)mrow * HIDDEN + ka + 1];
    float p0 = PW[ka], p1 = PW[ka + 1];
    v2f a, b;
    a.x = h0 > 0.f ? h0 : h0 * p0;     // fused PReLU
    a.y = h1v > 0.f ? h1v : h1v * p1;
    b.x = B[ka * OUT_FEAT + n0 + r];
    b.y = B[(ka + 1) * OUT_FEAT + n0 + r];
    c = __builtin_amdgcn_wmma_f32_16x16x4_f32(false, a, false, b, (short)0, c,
                                              false, false);
  }
#pragma unroll
  for (int v = 0; v < 8; ++v) {
    int row = m0 + v + 8 * half;
    if (row < Nn) XL[(size_t)row * OUT_FEAT + n0 + r] = c[v];
  }
}

// ---------- per-node attention dots: a_s = xl . att_src, a_d = xl . att_dst ----------
__global__ void att_kernel(const float* __restrict__ xl,
                           const float* __restrict__ ws,
                           const float* __restrict__ wd,
                           float* a_s, float* a_d, int Nn) {
  int wid = (blockIdx.x * blockDim.x + threadIdx.x) >> 5;  // one wave per node
  int lane = threadIdx.x & 31;
  if (wid >= Nn) return;
  float2 x = *(const float2*)(xl + (size_t)wid * OUT_FEAT + lane * 2);
  float2 s = *(const float2*)(ws + lane * 2);
  float2 d = *(const float2*)(wd + lane * 2);
  float ps = x.x * s.x + x.y * s.y;
  float pd = x.x * d.x + x.y * d.y;
#pragma unroll
  for (int off = 16; off; off >>= 1) {
    ps += __shfl_down(ps, off);
    pd += __shfl_down(pd, off);
  }
  if (lane == 0) { a_s[wid] = ps; a_d[wid] = pd; }
}

// ---------- GAT edge passes ----------
__global__ void edge_max_kernel(const int* __restrict__ src, const int* __restrict__ dst,
                                const float* __restrict__ a_s, const float* __restrict__ a_d,
                                unsigned* menc, int E, int Nn) {
  int e = blockIdx.x * blockDim.x + threadIdx.x;
  if (e >= E + Nn) return;
  int s, d;
  if (e < E) { s = src[e]; d = dst[e]; } else { s = d = e - E; }
  float v = lrelu02(a_s[s] + a_d[d]);
  atomicMax(&menc[d], enc_f32(v));
}

__global__ void edge_denom_kernel(const int* __restrict__ src, const int* __restrict__ dst,
                                  const float* __restrict__ a_s, const float* __restrict__ a_d,
                                  const unsigned* __restrict__ menc, float* denom,
                                  int E, int Nn) {
  int e = blockIdx.x * blockDim.x + threadIdx.x;
  if (e >= E + Nn) return;
  int s, d;
  if (e < E) { s = src[e]; d = dst[e]; } else { s = d = e - E; }
  float v = lrelu02(a_s[s] + a_d[d]);
  atomicAdd(&denom[d], expf(v - dec_f32(menc[d])));
}

__global__ void gat_scatter_kernel(const int* __restrict__ src, const int* __restrict__ dst,
                                   const float* __restrict__ a_s, const float* __restrict__ a_d,
                                   const unsigned* __restrict__ menc,
                                   const float* __restrict__ denom,
                                   const float* __restrict__ xl,
                                   float* out, int E, int Nn) {
  int wid = (blockIdx.x * blockDim.x + threadIdx.x) >> 5;  // one wave per edge
  int lane = threadIdx.x & 31;
  if (wid >= E + Nn) return;
  int s, d;
  if (wid < E) { s = src[wid]; d = dst[wid]; } else { s = d = wid - E; }
  float v = lrelu02(a_s[s] + a_d[d]);
  float alpha = expf(v - dec_f32(menc[d])) / denom[d];
  float2 x = *(const float2*)(xl + (size_t)s * OUT_FEAT + lane * 2);
  float* o = out + (size_t)d * OUT_FEAT + lane * 2;
  atomicAdd(o + 0, alpha * x.x);
  atomicAdd(o + 1, alpha * x.y);
}

extern "C" void kernel_launch(void* const* d_in, const int* in_sizes, int n_in,
                              void* d_out, int out_size, void* d_ws, size_t ws_size,
                              hipStream_t stream) {
  (void)n_in; (void)out_size; (void)ws_size;
  const float* z    = (const float*)d_in[0];
  const int*   ei   = (const int*)d_in[1];
  const float* w1   = (const float*)d_in[2];
  const float* b1   = (const float*)d_in[3];
  const float* pw   = (const float*)d_in[4];
  const float* w2   = (const float*)d_in[5];
  const float* asrc = (const float*)d_in[6];
  const float* adst = (const float*)d_in[7];
  const float* b2   = (const float*)d_in[8];
  float* out = (float*)d_out;

  const int N = in_sizes[0] / IN_FEAT;
  const int E = in_sizes[1] / 2;
  const int* esrc = ei;
  const int* edst = ei + E;
  const int ET = E + N;

  // workspace carve-out (256B aligned)
  char* wp = (char*)d_ws;
  auto take = [&](size_t bytes) {
    char* p = wp; wp += (bytes + 255) & ~(size_t)255; return p;
  };
  float*    hx    = (float*)take((size_t)N * HIDDEN * sizeof(float));
  float*    h1    = (float*)take((size_t)N * HIDDEN * sizeof(float));
  float*    xl    = (float*)take((size_t)N * OUT_FEAT * sizeof(float));
  float*    a_s   = (float*)take((size_t)N * sizeof(float));
  float*    a_d   = (float*)take((size_t)N * sizeof(float));
  float*    deg   = (float*)take((size_t)N * sizeof(float));
  float*    dinv  = (float*)take((size_t)N * sizeof(float));
  unsigned* menc  = (unsigned*)take((size_t)N * sizeof(unsigned));
  float*    denom = (float*)take((size_t)N * sizeof(float));

  const int T = 256;
  // init
  init_nodes_kernel<<<(N + T - 1) / T, T, 0, stream>>>(deg, menc, denom, N);
  init_bias_kernel<<<(N * HIDDEN + T - 1) / T, T, 0, stream>>>(h1, b1, N * HIDDEN, HIDDEN - 1);
  init_bias_kernel<<<(N * OUT_FEAT + T - 1) / T, T, 0, stream>>>(out, b2, N * OUT_FEAT, OUT_FEAT - 1);
  // GCN degrees
  deg_kernel<<<(E + T - 1) / T, T, 0, stream>>>(edst, deg, E);
  dinv_kernel<<<(N + T - 1) / T, T, 0, stream>>>(deg, dinv, N);
  // GCN: linear + scatter (self loops virtual), bias pre-seeded in h1
  gemm1_wmma_kernel<<<(N + 15) / 16, 256, 0, stream>>>(z, w1, hx, N);
  gcn_scatter_kernel<<<(ET * 32 + T - 1) / T, T, 0, stream>>>(esrc, edst, dinv, hx, h1, E, N);
  // GAT: PReLU fused into GEMM2 A-loads
  gemm2_wmma_kernel<<<(N + 15) / 16, 128, 0, stream>>>(h1, pw, w2, xl, N);
  att_kernel<<<(N * 32 + T - 1) / T, T, 0, stream>>>(xl, asrc, adst, a_s, a_d, N);
  edge_max_kernel<<<(ET + T - 1) / T, T, 0, stream>>>(esrc, edst, a_s, a_d, menc, E, N);
  edge_denom_kernel<<<(ET + T - 1) / T, T, 0, stream>>>(esrc, edst, a_s, a_d, menc, denom, E, N);
  gat_scatter_kernel<<<(ET * 32 + T - 1) / T, T, 0, stream>>>(esrc, edst, a_s, a_d, menc,
                                                             denom, xl, out, E, N);
}